// SequentialTask_9543417332175
// MI455X (gfx1250) — compile-verified
//
#include <hip/hip_runtime.h>
#include <hip/hip_bf16.h>

// ---------------------------------------------------------------------------
// Sampled embedding-dot + BCE loss for MI455X (gfx1250, wave32).
//
//   final[p]  = dot(flat[r], W0[c][0:768]) + W0[c][768]          (part 0)
//   final1[p] = dot(flat[r][576:768], W1[c][0:192]) + W1[c][192] (part 1)
//   loss      = sum( max(x,0) - x*y + log1p(exp(-|x|)) )
//
// Strategy: one wave handles 16 pairs; V_WMMA_F32_16X16X4_F32 accumulates
// D = A(16x4) x B(4x16); the 16 needed dot products are the diagonal of the
// 16x16 accumulator. The WMMA A/B VGPR layouts mean every lane feeds the
// matrix pipe with exactly one 8-byte gather from its pair's feature row and
// one from its weight row per K-step of 4 — optimal for this L2-gather-bound
// workload (all tables fit in the 192 MB L2).
// ---------------------------------------------------------------------------

typedef float v2f __attribute__((ext_vector_type(2)));
typedef float v8f __attribute__((ext_vector_type(8)));

#define NPAIRS   300000
#define NTILES   (NPAIRS / 16)      // 18750 tiles per pair set
#define WAVES_PB 8                  // 256 threads = 8 waves per block

// 8-byte load that is legal at 4-byte alignment (weight row stride is 769/193
// floats, so rows are only 4-byte aligned).
__device__ __forceinline__ v2f ld2(const float* p) {
    v2f v;
    __builtin_memcpy(&v, p, 8);
    return v;
}

__global__ __launch_bounds__(256) void pair_dot_wmma(
    const float* __restrict__ flat,      // (8192, 768) features
    const float* __restrict__ wtab,      // (30000, wstride) weights (last col = bias)
    const long long* __restrict__ idx,   // (NPAIRS, 2) int64 [row, code]
    const float* __restrict__ yv,        // (NPAIRS,) targets
    float* __restrict__ out_final,       // (NPAIRS,) or nullptr
    float* __restrict__ loss_partial,    // (NTILES,) per-wave partial loss
    int K, int wstride, int foff, int ntiles)
{
    const int tile = blockIdx.x * WAVES_PB + (threadIdx.x >> 5);
    if (tile >= ntiles) return;               // wave-uniform exit; EXEC stays full

    const int lane = threadIdx.x & 31;
    const int m    = lane & 15;               // pair within tile
    const int half = lane >> 4;               // K sub-offset selector (0 -> k, 1 -> k+2)
    const int p    = tile * 16 + m;

    const long long r = idx[2 * (long long)p + 0];
    const long long c = idx[2 * (long long)p + 1];

    const float* fb = flat + (size_t)r * 768 + foff + 2 * half;
    const float* wb = wtab + (size_t)c * wstride + 2 * half;

    v8f acc = {};
    // A(16x4): lane m holds f[r_m][k..k+1], lane m+16 holds f[r_m][k+2..k+3].
    // B(4x16): lane n holds w[c_n][k..k+1], lane n+16 holds w[c_n][k+2..k+3].
    // => identical per-lane gather pattern for both operands.
    for (int k = 0; k < K; k += 4) {
        v2f a = ld2(fb + k);
        v2f b = ld2(wb + k);
        acc = __builtin_amdgcn_wmma_f32_16x16x4_f32(
            /*neg_a=*/false, a, /*neg_b=*/false, b,
            /*c_mod=*/(short)0, acc, /*reuse_a=*/false, /*reuse_b=*/false);
    }

    // C/D layout: VGPR v, lanes 0-15 -> (M=v, N=lane); lanes 16-31 -> (M=v+8, N=lane-16).
    // Diagonal (m,m): m<8 lives at lane m, acc[m]; m>=8 lives at lane m+16, acc[m-8].
    float val = 0.0f;
#pragma unroll
    for (int v = 0; v < 8; ++v)
        val += ((lane & 7) == v) ? acc[v] : 0.0f;
    // Route diag(m) to lane m (lanes 0..15): source lane is m (m<8) or m+16 (m>=8).
    const int src = (lane < 8) ? lane : (lane + 16);
    val = __shfl(val, src, 32);

    float term = 0.0f;
    if (lane < 16) {
        val += wtab[(size_t)c * wstride + (wstride - 1)];   // bias (feature bias col == 1)
        if (out_final) out_final[p] = val;
        const float t = yv[p];
        // logaddexp(0, x) - x*t, numerically stable
        term = fmaxf(val, 0.0f) - val * t + log1pf(__expf(-fabsf(val)));
    }

    // Wave reduction of the 16 loss terms (lanes 16-31 hold 0).
#pragma unroll
    for (int off = 16; off > 0; off >>= 1)
        term += __shfl_down(term, off, 32);
    if (lane == 0) loss_partial[tile] = term;
}

// Deterministic fixed-order reduction of all per-wave partials -> scalar loss.
__global__ __launch_bounds__(256) void reduce_loss(
    const float* __restrict__ partials, int n, float* __restrict__ out)
{
    __shared__ float sm[256];
    float s = 0.0f;
    for (int i = threadIdx.x; i < n; i += 256) s += partials[i];
    sm[threadIdx.x] = s;
    __syncthreads();
#pragma unroll
    for (int st = 128; st > 0; st >>= 1) {
        if ((int)threadIdx.x < st) sm[threadIdx.x] += sm[threadIdx.x + st];
        __syncthreads();
    }
    if (threadIdx.x == 0) out[0] = sm[0];
}

extern "C" void kernel_launch(void* const* d_in, const int* in_sizes, int n_in,
                              void* d_out, int out_size, void* d_ws, size_t ws_size,
                              hipStream_t stream)
{
    const float*     flat = (const float*)d_in[0];      // rnn_output (16,512,768)
    const float*     w0   = (const float*)d_in[1];      // weight  (30000, 769)
    const float*     w1   = (const float*)d_in[2];      // weight1 (30000, 193)
    const long long* idx0 = (const long long*)d_in[3];  // (300000, 2) int64
    const float*     y0   = (const float*)d_in[4];
    // d_in[5] = seen0 (unused by reference)
    const long long* idx1 = (const long long*)d_in[6];
    const float*     y1   = (const float*)d_in[7];
    // d_in[8] = seen1 (unused by reference)

    float* out = (float*)d_out;          // [0..299999] = final, [300000] = loss
    float* ws  = (float*)d_ws;           // 2*NTILES partial loss sums (150 KB)

    const int grid = (NTILES + WAVES_PB - 1) / WAVES_PB;

    // Part 0: K=768, stride 769, features start at col 0, writes `final`.
    pair_dot_wmma<<<grid, 256, 0, stream>>>(
        flat, w0, idx0, y0, out, ws, 768, 769, 0, NTILES);

    // Part 1: K=192, stride 193, features start at col 576, loss only.
    pair_dot_wmma<<<grid, 256, 0, stream>>>(
        flat, w1, idx1, y1, nullptr, ws + NTILES, 192, 193, 576, NTILES);

    // Deterministic final sum -> d_out[300000].
    reduce_loss<<<1, 256, 0, stream>>>(ws, 2 * NTILES, out + NPAIRS);
}